// GCNHeatmapModel_33363305955792
// MI455X (gfx1250) — compile-verified
//
#include <hip/hip_runtime.h>
#include <math.h>

#define N_NODES 768
#define N_EDGES 12288
#define DIN 6
#define H 256

typedef float v2f __attribute__((ext_vector_type(2)));
typedef float v8f __attribute__((ext_vector_type(8)));

// ---------------- degree / normalization ----------------
__global__ void init_deg_kernel(int* __restrict__ deg) {
    int v = blockIdx.x * blockDim.x + threadIdx.x;
    if (v < N_NODES) deg[v] = 1;  // self loop
}

__global__ void count_deg_kernel(const int* __restrict__ dst, int* __restrict__ deg) {
    int e = blockIdx.x * blockDim.x + threadIdx.x;
    if (e < N_EDGES) atomicAdd(&deg[dst[e]], 1);
}

__global__ void dinv_kernel(const int* __restrict__ deg, float* __restrict__ dinv) {
    int v = blockIdx.x * blockDim.x + threadIdx.x;
    if (v < N_NODES) dinv[v] = rsqrtf((float)deg[v]);
}

// ---------------- x @ Wc1 (K=6, VALU) ----------------
__global__ void xw_kernel(const float* __restrict__ x, const float* __restrict__ W,
                          float* __restrict__ h0) {
    int n = blockIdx.x, c = threadIdx.x;
    float acc = 0.f;
#pragma unroll
    for (int k = 0; k < DIN; ++k) acc = fmaf(x[n * DIN + k], W[k * H + c], acc);
    h0[n * H + c] = acc;
}

// out[n,c] = dinv[n]^2 * hin[n,c] + bias[c]   (self-loop term + bias)
__global__ void self_bias_kernel(const float* __restrict__ hin, const float* __restrict__ dinv,
                                 const float* __restrict__ bias, float* __restrict__ out) {
    int n = blockIdx.x, c = threadIdx.x;
    float di = dinv[n];
    out[n * H + c] = fmaf(di * di, hin[n * H + c], bias[c]);
}

// out[dst,c] += dinv[src]*dinv[dst]*hin[src,c]  (one block per edge)
__global__ void edge_agg_kernel(const int* __restrict__ src, const int* __restrict__ dst,
                                const float* __restrict__ dinv, const float* __restrict__ hin,
                                float* __restrict__ out) {
    int e = blockIdx.x;
    int s = src[e], d = dst[e];
    float w = dinv[s] * dinv[d];
    int c = threadIdx.x;
    unsafeAtomicAdd(&out[d * H + c], w * hin[s * H + c]);
}

__global__ void elu_kernel(float* __restrict__ h) {
    int i = blockIdx.x * blockDim.x + threadIdx.x;
    float v = h[i];
    h[i] = v > 0.f ? v : expm1f(v);
}

// ---------------- C[768,256] = A[768,256] @ B[256,256], f32 WMMA ----------------
// One wave per 16x16 tile, K-loop of V_WMMA_F32_16X16X4_F32. 8 waves/block.
__global__ void wmma_gemm_kernel(const float* __restrict__ A, const float* __restrict__ B,
                                 float* __restrict__ C) {
    int wave = threadIdx.x >> 5;
    int lane = threadIdx.x & 31;
    int tile = blockIdx.x * 8 + wave;          // 768 tiles total
    int tm = tile >> 4;                        // 48 M-tiles
    int tn = tile & 15;                        // 16 N-tiles
    int half = lane >> 4;
    int l16  = lane & 15;

    const float* arow = A + (tm * 16 + l16) * H;   // A: lane row, K varies
    const float* bcol = B + tn * 16 + l16;         // B: lane col, K varies

    v8f acc = {0.f, 0.f, 0.f, 0.f, 0.f, 0.f, 0.f, 0.f};
    for (int k = 0; k < H; k += 4) {
        int ka = k + 2 * half;                 // half-wave K split
        v2f a;
        a.x = arow[ka];
        a.y = arow[ka + 1];
        v2f b;
        b.x = bcol[ka * H];
        b.y = bcol[(ka + 1) * H];
        acc = __builtin_amdgcn_wmma_f32_16x16x4_f32(
            false, a, false, b, (short)0, acc, false, false);
    }
    // D layout: VGPR r -> M = r + 8*half, N = l16
    float* crow = C + (tm * 16 + half * 8) * H + tn * 16 + l16;
#pragma unroll
    for (int r = 0; r < 8; ++r) crow[r * H] = acc[r];
}

// ---------------- pairwise relu-dot: logits[i,j] ----------------
__global__ void logits_kernel(const float* __restrict__ ai, const float* __restrict__ aj,
                              const float* __restrict__ be1, const float* __restrict__ We2,
                              const float* __restrict__ be2, float* __restrict__ out) {
    __shared__ float ais[16 * 257];  // pad to avoid 64-bank conflicts
    __shared__ float ajs[16 * 257];
    __shared__ float b1s[H];
    __shared__ float w2s[H];
    int tid = threadIdx.x;
    int bi = blockIdx.x, bj = blockIdx.y;
#pragma unroll
    for (int t = 0; t < 16; ++t) {
        int idx = tid + t * 256;
        int r = idx >> 8, c = idx & 255;
        ais[r * 257 + c] = ai[(bi * 16 + r) * H + c];
        ajs[r * 257 + c] = aj[(bj * 16 + r) * H + c];
    }
    b1s[tid] = be1[tid];
    w2s[tid] = We2[tid];
    __syncthreads();

    int ti = tid >> 4, tj = tid & 15;
    const float* pa = &ais[ti * 257];
    const float* pb = &ajs[tj * 257];
    float acc = be2[0];
    for (int h = 0; h < H; ++h) {
        float v = pa[h] + pb[h] + b1s[h];
        acc = fmaf(fmaxf(v, 0.f), w2s[h], acc);
    }
    out[(bi * 16 + ti) * N_NODES + (bj * 16 + tj)] = acc;  // TEMPERATURE == 1
}

// ---------------- row softmax with masked diagonal ----------------
__global__ void softmax_kernel(float* __restrict__ out) {
    __shared__ float red[256];
    int i = blockIdx.x, tid = threadIdx.x;
    float* row = out + i * N_NODES;

    float m = -3.402823466e38f;
    for (int j = tid; j < N_NODES; j += 256)
        if (j != i) m = fmaxf(m, row[j]);
    red[tid] = m;
    __syncthreads();
    for (int s = 128; s > 0; s >>= 1) {
        if (tid < s) red[tid] = fmaxf(red[tid], red[tid + s]);
        __syncthreads();
    }
    m = red[0];
    __syncthreads();

    float sum = 0.f;
    for (int j = tid; j < N_NODES; j += 256)
        if (j != i) sum += expf(row[j] - m);
    red[tid] = sum;
    __syncthreads();
    for (int s = 128; s > 0; s >>= 1) {
        if (tid < s) red[tid] += red[tid + s];
        __syncthreads();
    }
    sum = red[0];
    __syncthreads();

    float inv = 1.f / sum;
    for (int j = tid; j < N_NODES; j += 256)
        row[j] = (j == i) ? 0.f : expf(row[j] - m) * inv;
}

extern "C" void kernel_launch(void* const* d_in, const int* in_sizes, int n_in,
                              void* d_out, int out_size, void* d_ws, size_t ws_size,
                              hipStream_t stream) {
    const float* x   = (const float*)d_in[0];
    const int*   ei  = (const int*)d_in[1];
    const float* Wc1 = (const float*)d_in[2];
    const float* bc1 = (const float*)d_in[3];
    const float* Wc2 = (const float*)d_in[4];
    const float* bc2 = (const float*)d_in[5];
    const float* We1 = (const float*)d_in[6];
    const float* be1 = (const float*)d_in[7];
    const float* We2 = (const float*)d_in[8];
    const float* be2 = (const float*)d_in[9];
    float* out = (float*)d_out;

    // workspace layout (~4.7 MB)
    char* ws = (char*)d_ws;
    int*   deg  = (int*)ws;                          // 768 ints
    float* dinv = (float*)(ws + N_NODES * 4);        // 768 floats
    float* h0   = (float*)(ws + 2 * N_NODES * 4);    // 768*256 each below
    float* h1   = h0 + N_NODES * H;
    float* t2   = h1 + N_NODES * H;
    float* h2   = t2 + N_NODES * H;
    float* ai   = h2 + N_NODES * H;
    float* aj   = ai + N_NODES * H;

    const int* src = ei;
    const int* dst = ei + N_EDGES;

    // degree + norm
    init_deg_kernel<<<3, 256, 0, stream>>>(deg);
    count_deg_kernel<<<N_EDGES / 256, 256, 0, stream>>>(dst, deg);
    dinv_kernel<<<3, 256, 0, stream>>>(deg, dinv);

    // GCN layer 1
    xw_kernel<<<N_NODES, 256, 0, stream>>>(x, Wc1, h0);
    self_bias_kernel<<<N_NODES, 256, 0, stream>>>(h0, dinv, bc1, h1);
    edge_agg_kernel<<<N_EDGES, 256, 0, stream>>>(src, dst, dinv, h0, h1);
    elu_kernel<<<N_NODES, 256, 0, stream>>>(h1);

    // GCN layer 2
    wmma_gemm_kernel<<<96, 256, 0, stream>>>(h1, Wc2, t2);
    self_bias_kernel<<<N_NODES, 256, 0, stream>>>(t2, dinv, bc2, h2);
    edge_agg_kernel<<<N_EDGES, 256, 0, stream>>>(src, dst, dinv, t2, h2);

    // edge scorer projections
    wmma_gemm_kernel<<<96, 256, 0, stream>>>(h2, We1, ai);
    wmma_gemm_kernel<<<96, 256, 0, stream>>>(h2, We1 + H * H, aj);

    // pairwise relu-dot + softmax
    logits_kernel<<<dim3(48, 48), 256, 0, stream>>>(ai, aj, be1, We2, be2, out);
    softmax_kernel<<<N_NODES, 256, 0, stream>>>(out);
}